// TAOBAOGAT_35132832481404
// MI455X (gfx1250) — compile-verified
//
#include <hip/hip_runtime.h>
#include <math.h>

typedef __attribute__((ext_vector_type(2))) float v2f;
typedef __attribute__((ext_vector_type(8))) float v8f;

#define NEG_SLOPE 0.2f

// ---------------------------------------------------------------------------
// Float atomic max via sign-aware integer atomics (race-safe with -inf init).
// ---------------------------------------------------------------------------
__device__ __forceinline__ void atomicMaxFloat(float* addr, float val) {
    if (val >= 0.0f)
        atomicMax((int*)addr, __float_as_int(val));
    else
        atomicMin((unsigned int*)addr, __float_as_uint(val));
}

// ---------------------------------------------------------------------------
// Dual-output fp32 WMMA GEMM: Cl = A @ Bl, Cr = A @ Br.
// A: [M,K] row-major, B: [K,H] row-major, C: [M,H].
// One wave computes one 16x16 tile of BOTH outputs (A fragment reused).
// V_WMMA_F32_16X16X4_F32 operand layouts (ISA 7.12.2):
//   A 16x4 : lanes 0-15 rows M, v.x/v.y = K{0,1}; lanes 16-31 = K{2,3}
//   B 4x16 : lanes 0-15 cols N with v.x/v.y = K{0,1}; lanes 16-31 = K{2,3}
//   C/D    : VGPR r: lanes 0-15 -> M=r, lanes 16-31 -> M=r+8, N=lane&15
// ---------------------------------------------------------------------------
__global__ void dual_gemm_wmma_f32(const float* __restrict__ A,
                                   const float* __restrict__ Bl,
                                   const float* __restrict__ Br,
                                   float* __restrict__ Cl,
                                   float* __restrict__ Cr,
                                   int M, int K, int H) {
    const int tilesH = H >> 4;
    const int totalTiles = (M >> 4) * tilesH;
    const int wave = (int)((blockIdx.x * blockDim.x + threadIdx.x) >> 5);
    if (wave >= totalTiles) return;           // uniform per-wave: EXEC stays all-1s

    const int tm   = wave / tilesH;
    const int tn   = wave - tm * tilesH;
    const int lane = threadIdx.x & 31;
    const int half = lane >> 4;               // 0: K pair {0,1}, 1: K pair {2,3}
    const int lm   = lane & 15;

    const int rowA = (tm << 4) + lm;
    const int colB = (tn << 4) + lm;
    const float* __restrict__ arow = A + (size_t)rowA * K;

    v8f accl = {};
    v8f accr = {};

    for (int k = 0; k < K; k += 4) {
        const int ka = k + (half << 1);
        v2f a, bl, br;
        a.x  = arow[ka];
        a.y  = arow[ka + 1];
        bl.x = Bl[(size_t)ka * H + colB];
        bl.y = Bl[(size_t)(ka + 1) * H + colB];
        br.x = Br[(size_t)ka * H + colB];
        br.y = Br[(size_t)(ka + 1) * H + colB];
        accl = __builtin_amdgcn_wmma_f32_16x16x4_f32(false, a, false, bl,
                                                     (short)0, accl, false, false);
        accr = __builtin_amdgcn_wmma_f32_16x16x4_f32(false, a, false, br,
                                                     (short)0, accr, false, false);
    }

    const int baseRow = (tm << 4) + (half << 3);
#pragma unroll
    for (int r = 0; r < 8; ++r) {
        Cl[(size_t)(baseRow + r) * H + colB] = accl[r];
        Cr[(size_t)(baseRow + r) * H + colB] = accr[r];
    }
}

// ---------------------------------------------------------------------------
// Per-layer init: m = -inf, denom = 0, h = 0
// ---------------------------------------------------------------------------
__global__ void init_layer(float* __restrict__ m, float* __restrict__ den,
                           float* __restrict__ h, int N, int H) {
    const int stride = gridDim.x * blockDim.x;
    const size_t total = (size_t)N * H;
    for (size_t t = blockIdx.x * blockDim.x + threadIdx.x; t < total; t += stride)
        h[t] = 0.0f;
    for (int t = blockIdx.x * blockDim.x + threadIdx.x; t < N; t += stride) {
        m[t]   = -INFINITY;
        den[t] = 0.0f;
    }
}

// ---------------------------------------------------------------------------
// Wave per edge: e[i] = att . leaky_relu(xl[src] + xr[dst]); atomic-max m[dst]
// Edges i in [0,E) come from edge_index; i in [E, E+N) are self-loops.
// ---------------------------------------------------------------------------
__global__ void edge_logits(const float* __restrict__ xl,
                            const float* __restrict__ xr,
                            const float* __restrict__ att,
                            const int* __restrict__ ei,
                            int E, int N, int H,
                            float* __restrict__ e, float* __restrict__ m) {
    const int lane  = threadIdx.x & 31;
    const int wid   = (int)((blockIdx.x * blockDim.x + threadIdx.x) >> 5);
    const int nwave = (int)((gridDim.x * blockDim.x) >> 5);
    const int total = E + N;

    for (int i = wid; i < total; i += nwave) {
        int s, d;
        if (i < E) { s = ei[i]; d = ei[E + i]; }
        else       { s = i - E; d = s; }

        // prefetch next iteration's gather rows into cache
        const int inext = i + nwave;
        if (inext < total) {
            const int sn = (inext < E) ? ei[inext]     : inext - E;
            const int dn = (inext < E) ? ei[E + inext] : inext - E;
            __builtin_prefetch(xl + (size_t)sn * H, 0, 3);
            __builtin_prefetch(xr + (size_t)dn * H, 0, 3);
        }

        float part = 0.0f;
        for (int c = lane; c < H; c += 32) {
            float v = xl[(size_t)s * H + c] + xr[(size_t)d * H + c];
            v = (v > 0.0f) ? v : NEG_SLOPE * v;
            part += v * att[c];
        }
#pragma unroll
        for (int off = 16; off > 0; off >>= 1)
            part += __shfl_xor(part, off, 32);

        if (lane == 0) {
            e[i] = part;
            atomicMaxFloat(m + d, part);
        }
    }
}

// ---------------------------------------------------------------------------
// Thread per edge: ex = exp(e - m[dst]); e[i] = ex; denom[dst] += ex
// ---------------------------------------------------------------------------
__global__ void edge_exp(float* __restrict__ e, const float* __restrict__ m,
                         float* __restrict__ den, const int* __restrict__ ei,
                         int E, int N) {
    const int stride = gridDim.x * blockDim.x;
    const int total  = E + N;
    for (int i = blockIdx.x * blockDim.x + threadIdx.x; i < total; i += stride) {
        const int d  = (i < E) ? ei[E + i] : i - E;
        const float ex = __expf(e[i] - m[d]);
        e[i] = ex;
        atomicAdd(den + d, ex);
    }
}

// ---------------------------------------------------------------------------
// Wave per edge: h[dst] += alpha * xl[src], alpha = e[i] / (denom[dst]+eps)
// ---------------------------------------------------------------------------
__global__ void edge_scatter(const float* __restrict__ xl,
                             const float* __restrict__ e,
                             const float* __restrict__ den,
                             const int* __restrict__ ei,
                             int E, int N, int H, float* __restrict__ h) {
    const int lane  = threadIdx.x & 31;
    const int wid   = (int)((blockIdx.x * blockDim.x + threadIdx.x) >> 5);
    const int nwave = (int)((gridDim.x * blockDim.x) >> 5);
    const int total = E + N;

    for (int i = wid; i < total; i += nwave) {
        int s, d;
        if (i < E) { s = ei[i]; d = ei[E + i]; }
        else       { s = i - E; d = s; }

        const int inext = i + nwave;
        if (inext < total) {
            const int sn = (inext < E) ? ei[inext] : inext - E;
            __builtin_prefetch(xl + (size_t)sn * H, 0, 3);
        }

        const float alpha = e[i] / (den[d] + 1e-16f);
        for (int c = lane; c < H; c += 32)
            atomicAdd(h + (size_t)d * H + c, alpha * xl[(size_t)s * H + c]);
    }
}

// ---------------------------------------------------------------------------
// h = relu(h + b)
// ---------------------------------------------------------------------------
__global__ void bias_relu(float* __restrict__ h, const float* __restrict__ b,
                          int N, int H) {
    const int stride = gridDim.x * blockDim.x;
    const size_t total = (size_t)N * H;
    for (size_t t = blockIdx.x * blockDim.x + threadIdx.x; t < total; t += stride) {
        const float v = h[t] + b[(int)(t % H)];
        h[t] = (v > 0.0f) ? v : 0.0f;
    }
}

// ---------------------------------------------------------------------------
// Wave per label edge:
// out[l] = sum_c h[s][c]*h[d][c]*(Wp[c][0]+Wp[c][1]) + (bp[0]+bp[1])
// ---------------------------------------------------------------------------
__global__ void predict(const float* __restrict__ h, const int* __restrict__ eli,
                        int L, int H, const float* __restrict__ Wp,
                        const float* __restrict__ bp, float* __restrict__ out) {
    const int lane  = threadIdx.x & 31;
    const int wid   = (int)((blockIdx.x * blockDim.x + threadIdx.x) >> 5);
    const int nwave = (int)((gridDim.x * blockDim.x) >> 5);
    const float bpsum = bp[0] + bp[1];

    for (int l = wid; l < L; l += nwave) {
        const int s = eli[l];
        const int d = eli[L + l];
        float part = 0.0f;
        for (int c = lane; c < H; c += 32) {
            const float w = Wp[2 * c] + Wp[2 * c + 1];
            part += h[(size_t)s * H + c] * h[(size_t)d * H + c] * w;
        }
#pragma unroll
        for (int off = 16; off > 0; off >>= 1)
            part += __shfl_xor(part, off, 32);
        if (lane == 0) out[l] = part + bpsum;
    }
}

// ---------------------------------------------------------------------------
// Host-side orchestration
// ---------------------------------------------------------------------------
extern "C" void kernel_launch(void* const* d_in, const int* in_sizes, int n_in,
                              void* d_out, int out_size, void* d_ws, size_t ws_size,
                              hipStream_t stream) {
    const float* x    = (const float*)d_in[0];
    const int*   ei   = (const int*)  d_in[1];
    const int*   eli  = (const int*)  d_in[2];
    const float* Wl1  = (const float*)d_in[3];
    const float* Wr1  = (const float*)d_in[4];
    const float* att1 = (const float*)d_in[5];
    const float* b1   = (const float*)d_in[6];
    const float* Wl2  = (const float*)d_in[7];
    const float* Wr2  = (const float*)d_in[8];
    const float* att2 = (const float*)d_in[9];
    const float* b2   = (const float*)d_in[10];
    const float* Wp   = (const float*)d_in[11];
    const float* bp   = (const float*)d_in[12];
    float* out = (float*)d_out;

    const int IN_C = 128, H1 = 96, H2 = 64;
    const int N = in_sizes[0] / IN_C;
    const int E = in_sizes[1] / 2;
    const int L = in_sizes[2] / 2;

    // workspace layout
    float* ws = (float*)d_ws;
    size_t o = 0;
    float* xl1 = ws + o; o += (size_t)N * H1;
    float* xr1 = ws + o; o += (size_t)N * H1;
    float* h1  = ws + o; o += (size_t)N * H1;
    float* xl2 = ws + o; o += (size_t)N * H2;
    float* xr2 = ws + o; o += (size_t)N * H2;
    float* h2  = ws + o; o += (size_t)N * H2;
    float* e   = ws + o; o += (size_t)(E + N);
    float* m   = ws + o; o += (size_t)N;
    float* den = ws + o; o += (size_t)N;

    const int BLK = 256;                       // 8 waves/block
    const int totalEdges = E + N;
    const int edgeWaveGrid   = 4096;           // wave-per-edge grid-stride
    const int edgeThreadGrid = (totalEdges + BLK - 1) / BLK;
    const int initGrid       = 1024;

    // ---- Layer 1 ----
    {
        const int tiles = (N >> 4) * (H1 >> 4);
        const int grid  = (tiles + (BLK >> 5) - 1) / (BLK >> 5);
        dual_gemm_wmma_f32<<<grid, BLK, 0, stream>>>(x, Wl1, Wr1, xl1, xr1, N, IN_C, H1);
    }
    init_layer<<<initGrid, BLK, 0, stream>>>(m, den, h1, N, H1);
    edge_logits<<<edgeWaveGrid, BLK, 0, stream>>>(xl1, xr1, att1, ei, E, N, H1, e, m);
    edge_exp<<<edgeThreadGrid, BLK, 0, stream>>>(e, m, den, ei, E, N);
    edge_scatter<<<edgeWaveGrid, BLK, 0, stream>>>(xl1, e, den, ei, E, N, H1, h1);
    {
        const int grid = (int)(((size_t)N * H1 + BLK - 1) / BLK);
        bias_relu<<<grid, BLK, 0, stream>>>(h1, b1, N, H1);
    }

    // ---- Layer 2 ----
    {
        const int tiles = (N >> 4) * (H2 >> 4);
        const int grid  = (tiles + (BLK >> 5) - 1) / (BLK >> 5);
        dual_gemm_wmma_f32<<<grid, BLK, 0, stream>>>(h1, Wl2, Wr2, xl2, xr2, N, H1, H2);
    }
    init_layer<<<initGrid, BLK, 0, stream>>>(m, den, h2, N, H2);
    edge_logits<<<edgeWaveGrid, BLK, 0, stream>>>(xl2, xr2, att2, ei, E, N, H2, e, m);
    edge_exp<<<edgeThreadGrid, BLK, 0, stream>>>(e, m, den, ei, E, N);
    edge_scatter<<<edgeWaveGrid, BLK, 0, stream>>>(xl2, e, den, ei, E, N, H2, h2);
    {
        const int grid = (int)(((size_t)N * H2 + BLK - 1) / BLK);
        bias_relu<<<grid, BLK, 0, stream>>>(h2, b2, N, H2);
    }

    // ---- Link prediction ----
    predict<<<4096, BLK, 0, stream>>>(h2, eli, L, H2, Wp, bp, out);
}